// MambaBlock_69295002353976
// MI455X (gfx1250) — compile-verified
//
#include <hip/hip_runtime.h>
#include <math.h>

// ---------------------------------------------------------------------------
// Mamba block forward for MI455X (gfx1250, wave32).
//   B=2, T=4096, D_MODEL=1024, D_INNER=2048, D_STATE=16, D_CONV=4
// Pipeline:
//   K1: xz = x @ W_in^T            (WMMA f32 16x16x4 GEMM, M=8192 N=4096 K=1024)
//   K2: depthwise causal conv + SiLU -> xb_act
//   K3: ssm = xb_act @ Wx^T  (N=33 skinny; block-per-token VALU reduce)
//   K4: selective scan fused with softplus(dt), Dp skip, SiLU(z) gate
//   K5: out = gated @ Wout^T       (WMMA f32 GEMM, M=8192 N=1024 K=2048)
// GEMMs use GLOBAL_LOAD_ASYNC_TO_LDS_B128 (ASYNCcnt) double-buffered staging
// when the builtin is available; otherwise a register-staged sync path.
// ---------------------------------------------------------------------------

typedef __attribute__((ext_vector_type(2))) float v2f;
typedef __attribute__((ext_vector_type(4))) float v4f;
typedef __attribute__((ext_vector_type(8))) float v8f;

#define BB      2
#define TT      4096
#define DMODEL  1024
#define DINNER  2048
#define DSTATE  16
#define NTOK    (BB * TT)          // 8192
#define NOUT33  (2 * DSTATE + 1)   // 33

#if defined(__gfx1250__) && \
    __has_builtin(__builtin_amdgcn_global_load_async_to_lds_b128) && \
    __has_builtin(__builtin_amdgcn_s_wait_asynccnt)
#define HAVE_ASYNC_LDS 1
#else
#define HAVE_ASYNC_LDS 0
#endif

// ------------------------------ GEMM (WMMA) --------------------------------
// C[M,N] = A[M,K] @ Bw[N,K]^T   (both row-major), fp32 accumulate.
// 256 threads = 8 waves arranged 4(M) x 2(N); wave tile 32x32 (4 accums).
// Workgroup tile 128x64. K staged through double-buffered LDS, chunk 32.
#define WG_M 128
#define WG_N 64
#define KC   32
#define LDSS 36   // LDS row stride (floats); 36*4=144 B keeps f4/f2 aligned

#if HAVE_ASYNC_LDS
// Builtin parameter type (from hipcc diagnostic): int __vector_size__(16)
// pointers in AS1 (global src) / AS3 (LDS dst).
typedef int b128_t __attribute__((__vector_size__(4 * sizeof(int))));
typedef __attribute__((address_space(1))) b128_t* g128p;
typedef __attribute__((address_space(3))) b128_t* l128p;
typedef __attribute__((address_space(1))) void*   g_vp;
typedef __attribute__((address_space(3))) void*   l_vp;

template <int NF4>
__device__ __forceinline__
void stage_async(const float* __restrict__ g, int ld, int row0, int kc,
                 float* lds, int tid)
{
    #pragma unroll
    for (int j = 0; j < NF4; ++j) {
        const int i   = tid + j * 256;
        const int row = i >> 3;
        const int kq  = (i & 7) << 2;
        g128p gp = (g128p)(g_vp)(float*)(g + (size_t)(row0 + row) * ld + kc + kq);
        l128p lp = (l128p)(l_vp)(lds + row * LDSS + kq);
        __builtin_amdgcn_global_load_async_to_lds_b128(gp, lp, 0, 0);
    }
}
#endif

__global__ __launch_bounds__(256)
void gemm_f32_wmma(const float* __restrict__ A, const float* __restrict__ Bw,
                   float* __restrict__ C, int M, int N, int K)
{
    __shared__ float Xs[2][WG_M * LDSS];
    __shared__ float Ws[2][WG_N * LDSS];

    const int tid  = threadIdx.x;
    const int m0   = blockIdx.x * WG_M;
    const int n0   = blockIdx.y * WG_N;
    const int w    = tid >> 5;        // wave id 0..7
    const int lane = tid & 31;
    const int wm   = w >> 1;          // 0..3 : wave row block (32 rows)
    const int wn   = w & 1;           // 0..1 : wave col block (32 cols)
    const int l15  = lane & 15;
    const int hh   = lane >> 4;       // lane half

    v8f acc00 = {}, acc01 = {}, acc10 = {}, acc11 = {};

    const int nch = K / KC;

#if HAVE_ASYNC_LDS
    // ---- async double-buffered pipeline (ASYNCcnt-tracked DMA into LDS) ----
    stage_async<4>(A, K, m0, 0, Xs[0], tid);
    stage_async<2>(Bw, K, n0, 0, Ws[0], tid);

    for (int c = 0; c < nch; ++c) {
        __builtin_amdgcn_s_wait_asynccnt(0);   // chunk c resident in LDS
        __syncthreads();                       // publish to all waves; frees other buf
        if (c + 1 < nch) {
            stage_async<4>(A, K, m0, (c + 1) * KC, Xs[(c + 1) & 1], tid);
            stage_async<2>(Bw, K, n0, (c + 1) * KC, Ws[(c + 1) & 1], tid);
        }
        const float* xs = Xs[c & 1];
        const float* ws = Ws[c & 1];
        #pragma unroll
        for (int kb = 0; kb < KC; kb += 4) {
            // A frag: element (m,k) -> lane m+16*(k>>1), vgpr k&1 => float2
            v2f a0 = *(const v2f*)(xs + (wm * 32      + l15) * LDSS + kb + 2 * hh);
            v2f a1 = *(const v2f*)(xs + (wm * 32 + 16 + l15) * LDSS + kb + 2 * hh);
            // B frag: element (k,n) -> lane n+16*(k>>1), vgpr k&1; B[k,n]=W[n,k]
            v2f b0 = *(const v2f*)(ws + (wn * 32      + l15) * LDSS + kb + 2 * hh);
            v2f b1 = *(const v2f*)(ws + (wn * 32 + 16 + l15) * LDSS + kb + 2 * hh);
            acc00 = __builtin_amdgcn_wmma_f32_16x16x4_f32(false, a0, false, b0, (short)0, acc00, false, false);
            acc01 = __builtin_amdgcn_wmma_f32_16x16x4_f32(false, a0, false, b1, (short)0, acc01, false, false);
            acc10 = __builtin_amdgcn_wmma_f32_16x16x4_f32(false, a1, false, b0, (short)0, acc10, false, false);
            acc11 = __builtin_amdgcn_wmma_f32_16x16x4_f32(false, a1, false, b1, (short)0, acc11, false, false);
        }
        __syncthreads();                       // compute(c) done before overwrite
    }
#else
    // ---- fallback: register-staged synchronous path (straight-line) ----
    for (int c = 0; c < nch; ++c) {
        const int kc = c * KC;
        v4f ra[4], rb[2];
        #pragma unroll
        for (int j = 0; j < 4; ++j) {
            const int i = tid + j * 256, row = i >> 3, kq = (i & 7) << 2;
            ra[j] = *(const v4f*)(A + (size_t)(m0 + row) * K + kc + kq);
        }
        #pragma unroll
        for (int j = 0; j < 2; ++j) {
            const int i = tid + j * 256, row = i >> 3, kq = (i & 7) << 2;
            rb[j] = *(const v4f*)(Bw + (size_t)(n0 + row) * K + kc + kq);
        }
        __syncthreads();
        #pragma unroll
        for (int j = 0; j < 4; ++j) {
            const int i = tid + j * 256, row = i >> 3, kq = (i & 7) << 2;
            *(v4f*)(&Xs[0][row * LDSS + kq]) = ra[j];
        }
        #pragma unroll
        for (int j = 0; j < 2; ++j) {
            const int i = tid + j * 256, row = i >> 3, kq = (i & 7) << 2;
            *(v4f*)(&Ws[0][row * LDSS + kq]) = rb[j];
        }
        __syncthreads();
        #pragma unroll
        for (int kb = 0; kb < KC; kb += 4) {
            v2f a0 = *(const v2f*)(&Xs[0][(wm * 32      + l15) * LDSS + kb + 2 * hh]);
            v2f a1 = *(const v2f*)(&Xs[0][(wm * 32 + 16 + l15) * LDSS + kb + 2 * hh]);
            v2f b0 = *(const v2f*)(&Ws[0][(wn * 32      + l15) * LDSS + kb + 2 * hh]);
            v2f b1 = *(const v2f*)(&Ws[0][(wn * 32 + 16 + l15) * LDSS + kb + 2 * hh]);
            acc00 = __builtin_amdgcn_wmma_f32_16x16x4_f32(false, a0, false, b0, (short)0, acc00, false, false);
            acc01 = __builtin_amdgcn_wmma_f32_16x16x4_f32(false, a0, false, b1, (short)0, acc01, false, false);
            acc10 = __builtin_amdgcn_wmma_f32_16x16x4_f32(false, a1, false, b0, (short)0, acc10, false, false);
            acc11 = __builtin_amdgcn_wmma_f32_16x16x4_f32(false, a1, false, b1, (short)0, acc11, false, false);
        }
    }
#endif

    // C/D layout: vgpr r -> rows r (lanes 0-15) / r+8 (lanes 16-31), N=lane&15
    #pragma unroll
    for (int r = 0; r < 8; ++r) {
        const int mA = m0 + wm * 32 + r + 8 * hh;
        const int mB = mA + 16;
        const int n  = n0 + wn * 32 + l15;
        C[(size_t)mA * N + n]      = acc00[r];
        C[(size_t)mA * N + n + 16] = acc01[r];
        C[(size_t)mB * N + n]      = acc10[r];
        C[(size_t)mB * N + n + 16] = acc11[r];
    }
}

// --------------------- depthwise causal conv + SiLU ------------------------
// xz layout: [NTOK, 2*DINNER]; cols [0,2048) = xb, [2048,4096) = z.
__global__ __launch_bounds__(256)
void conv_silu_kernel(const float* __restrict__ xz,
                      const float* __restrict__ conv_w,
                      const float* __restrict__ conv_b,
                      float* __restrict__ xb_act)
{
    const int i = blockIdx.x * 256 + threadIdx.x;   // over NTOK*DINNER
    if (i >= NTOK * DINNER) return;
    const int d  = i & (DINNER - 1);
    const int bt = i >> 11;                         // token index b*T+t
    const int t  = bt & (TT - 1);

    const float w0 = conv_w[d * 4 + 0];
    const float w1 = conv_w[d * 4 + 1];
    const float w2 = conv_w[d * 4 + 2];
    const float w3 = conv_w[d * 4 + 3];

    float acc = conv_b[d];
    if (t >= 3) acc += w0 * xz[(size_t)(bt - 3) * (2 * DINNER) + d];
    if (t >= 2) acc += w1 * xz[(size_t)(bt - 2) * (2 * DINNER) + d];
    if (t >= 1) acc += w2 * xz[(size_t)(bt - 1) * (2 * DINNER) + d];
    acc += w3 * xz[(size_t)bt * (2 * DINNER) + d];

    acc = acc / (1.0f + __expf(-acc));              // SiLU
    xb_act[(size_t)bt * DINNER + d] = acc;
}

// ---------------------- x-projection (N=33 skinny GEMM) --------------------
__global__ __launch_bounds__(256)
void xproj_kernel(const float* __restrict__ xb_act,
                  const float* __restrict__ Wx,
                  float* __restrict__ Bp, float* __restrict__ Cp,
                  float* __restrict__ dt_raw)
{
    const int tok  = blockIdx.x;
    const int tid  = threadIdx.x;
    const int lane = tid & 31;
    const int wv   = tid >> 5;

    const float* xr = xb_act + (size_t)tok * DINNER;

    float acc[NOUT33];
    #pragma unroll
    for (int o = 0; o < NOUT33; ++o) acc[o] = 0.0f;

    for (int k = tid; k < DINNER; k += 256) {
        const float xv = xr[k];
        #pragma unroll
        for (int o = 0; o < NOUT33; ++o)
            acc[o] += xv * Wx[o * DINNER + k];
    }

    #pragma unroll
    for (int o = 0; o < NOUT33; ++o) {
        #pragma unroll
        for (int off = 16; off > 0; off >>= 1)
            acc[o] += __shfl_xor(acc[o], off, 32);
    }

    __shared__ float red[NOUT33][8];
    if (lane == 0) {
        #pragma unroll
        for (int o = 0; o < NOUT33; ++o) red[o][wv] = acc[o];
    }
    __syncthreads();

    if (tid < NOUT33) {
        float s = 0.0f;
        #pragma unroll
        for (int j = 0; j < 8; ++j) s += red[tid][j];
        if (tid < DSTATE)            Bp[(size_t)tok * DSTATE + tid] = s;
        else if (tid < 2 * DSTATE)   Cp[(size_t)tok * DSTATE + (tid - DSTATE)] = s;
        else                         dt_raw[tok] = s;
    }
}

// --------------------------- selective scan + gate -------------------------
// One lane per (b, d, n): 65536 threads = 2048 wave32s. Group of 16 lanes
// shares (b,d); y_t = sum_n h_n * C_n via 16-wide shfl_xor. Writes gated
// in-place over xb_act (step-t loads precede the step-t store in-wave).
__global__ __launch_bounds__(256)
void scan_gate_kernel(const float* __restrict__ xz,       // for z gate
                      float* __restrict__ xbg,            // xb_act in, gated out
                      const float* __restrict__ Bp,
                      const float* __restrict__ Cp,
                      const float* __restrict__ dt_raw,
                      const float* __restrict__ Wdt,
                      const float* __restrict__ b_dt,
                      const float* __restrict__ A_log,
                      const float* __restrict__ Dp)
{
    const int tid = blockIdx.x * 256 + threadIdx.x;
    const int g   = tid >> 4;            // group = b*DINNER + d
    const int n   = tid & 15;            // state index
    const int b   = g >> 11;
    const int d   = g & (DINNER - 1);

    const float An  = -__expf(A_log[d * DSTATE + n]);
    const float wdt = Wdt[d];
    const float bdt = b_dt[d];
    const float dpd = Dp[d];

    float h = 0.0f;
    for (int t = 0; t < TT; ++t) {
        const int base = b * TT + t;
        const float x  = xbg[(size_t)base * DINNER + d];
        const float u  = dt_raw[base] * wdt + bdt;
        const float dt = (u > 20.0f) ? u : log1pf(__expf(u));   // softplus
        const float Bv = Bp[(size_t)base * DSTATE + n];
        const float Cv = Cp[(size_t)base * DSTATE + n];

        h = __expf(dt * An) * h + dt * Bv * x;

        float p = h * Cv;
        p += __shfl_xor(p, 8, 16);
        p += __shfl_xor(p, 4, 16);
        p += __shfl_xor(p, 2, 16);
        p += __shfl_xor(p, 1, 16);

        if (n == 0) {
            const float zv = xz[(size_t)base * (2 * DINNER) + DINNER + d];
            const float y  = p + dpd * x;
            xbg[(size_t)base * DINNER + d] = y * (zv / (1.0f + __expf(-zv)));
        }
    }
}

// ------------------------------- launcher ----------------------------------
extern "C" void kernel_launch(void* const* d_in, const int* in_sizes, int n_in,
                              void* d_out, int out_size, void* d_ws, size_t ws_size,
                              hipStream_t stream)
{
    const float* x      = (const float*)d_in[0];
    const float* W_in   = (const float*)d_in[1];
    const float* conv_w = (const float*)d_in[2];
    const float* conv_b = (const float*)d_in[3];
    const float* Wx     = (const float*)d_in[4];
    const float* Wdt    = (const float*)d_in[5];
    const float* b_dt   = (const float*)d_in[6];
    const float* A_log  = (const float*)d_in[7];
    const float* Dp     = (const float*)d_in[8];
    const float* Wout   = (const float*)d_in[9];
    float* out = (float*)d_out;

    // Workspace partition (floats): xz 33.55M, xb_act 16.78M (reused as gated),
    // Bp/Cp NTOK*16 each, dt_raw NTOK.  Total ~193 MB.
    float* ws     = (float*)d_ws;
    float* xz     = ws;
    float* xb_act = xz + (size_t)NTOK * 2 * DINNER;
    float* Bpb    = xb_act + (size_t)NTOK * DINNER;
    float* Cpb    = Bpb + (size_t)NTOK * DSTATE;
    float* dtr    = Cpb + (size_t)NTOK * DSTATE;

    dim3 blk(256);

    // K1: xz = x @ W_in^T   (M=8192, N=4096, K=1024)
    gemm_f32_wmma<<<dim3(NTOK / WG_M, (2 * DINNER) / WG_N), blk, 0, stream>>>(
        x, W_in, xz, NTOK, 2 * DINNER, DMODEL);

    // K2: depthwise conv + SiLU
    conv_silu_kernel<<<dim3((NTOK * DINNER) / 256), blk, 0, stream>>>(
        xz, conv_w, conv_b, xb_act);

    // K3: x-projection -> Bp, Cp, dt_raw
    xproj_kernel<<<dim3(NTOK), blk, 0, stream>>>(xb_act, Wx, Bpb, Cpb, dtr);

    // K4: selective scan + gate (writes gated in-place over xb_act)
    scan_gate_kernel<<<dim3((BB * DINNER * DSTATE) / 256), blk, 0, stream>>>(
        xz, xb_act, Bpb, Cpb, dtr, Wdt, b_dt, A_log, Dp);

    // K5: out = gated @ Wout^T  (M=8192, N=1024, K=2048)
    gemm_f32_wmma<<<dim3(NTOK / WG_M, DMODEL / WG_N), blk, 0, stream>>>(
        xb_act, Wout, out, NTOK, DMODEL, DINNER);
}